// DiffusionInteractionBlock_50311246905654
// MI455X (gfx1250) — compile-verified
//
#include <hip/hip_runtime.h>

// DiffusionInteractionBlock for MI455X (gfx1250), fp32 WMMA 16x16x4 path.
//
// Pipeline:
//   K1: node GEMMs  (node_scalars, s_up, v_up)          -- 5x WMMA streams
//   memset: zero msg accumulator (10000 x 1024 f32, L2-resident)
//   K2: fused edge MLP (265->128->128->512) + tensor-product messages
//       + segment-sum via global_atomic_add_f32
//   K3: output GEMMs (256->128, scalar + 3 vector comps), fused (n,128,4) store
//
// Workspace (floats): ns[10000*128] | s_up[10000*128] | v_up[10000*384] | msg[10000*1024]
//   total = 16,640,000 floats = 66.56 MB (assumed <= ws_size)

#define NN    10000
#define NE    160000
#define MULW  128
#define DIN   265      // 2*128 + 8 + 1
#define DPAD  268      // K extent for layer1 (multiple of 4, zero padded)
#define MLPS  276      // LDS stride for mlp_in tile (bank-friendly, 16B-aligned rows)
#define HSTR  130      // LDS stride for h tiles
#define NFS   516      // LDS stride for node_feats tile
#define INV_SQRT3 0.57735026918962576f

typedef float v2f __attribute__((ext_vector_type(2)));
typedef float v8f __attribute__((ext_vector_type(8)));

__device__ __forceinline__ v8f wmma4(v2f a, v2f b, v8f c) {
  // D = A(16x4,f32) x B(4x16,f32) + C(16x16,f32)
  return __builtin_amdgcn_wmma_f32_16x16x4_f32(false, a, false, b, (short)0, c,
                                               false, false);
}

__device__ __forceinline__ float siluf(float x) {
  return x * (1.0f / (1.0f + __expf(-x)));
}

// ---------------------------------------------------------------------------
// Kernel 1: node-level GEMMs.  grid = NN/16 blocks of 256 threads (8 waves).
// Wave w owns output columns [w*16, w*16+16).  A tile (16 nodes x 512 feats)
// staged in LDS once per block.
// ---------------------------------------------------------------------------
__global__ __launch_bounds__(256)
void k_node_up(const float* __restrict__ nf,
               const float* __restrict__ Wsc,
               const float* __restrict__ Wu0,
               const float* __restrict__ Wu1,
               float* __restrict__ ns,
               float* __restrict__ sup,
               float* __restrict__ vup) {
  __shared__ float lds[16 * NFS];
  const int tid = threadIdx.x;
  const int g   = blockIdx.x;            // node tile

  // cooperative stage: 16 rows x 512 floats (float4 moves, rows 16B aligned)
  {
    const float4* src = (const float4*)(nf + (size_t)g * 16 * 512);
    for (int i = tid; i < 16 * 128; i += 256) {
      const int row = i >> 7, c4 = i & 127;
      *(float4*)&lds[row * NFS + c4 * 4] = src[row * 128 + c4];
    }
  }
  __syncthreads();

  const int wv = tid >> 5, lane = tid & 31;
  const int half = lane >> 4, lm = lane & 15;
  const int col = wv * 16 + lm;

  v8f a_ns = {}, a_su = {}, a_v0 = {}, a_v1 = {}, a_v2 = {};

  for (int k = 0; k < 128; k += 4) {
    const int kr = k + 2 * half;
    const float* rowp = &lds[lm * NFS];
    // A fragments: scalar block + 3 vector components (stride-3 in memory)
    v2f as = *(const v2f*)&rowp[kr];
    v2f av0, av1, av2;
    av0.x = rowp[128 + kr * 3 + 0];  av0.y = rowp[128 + (kr + 1) * 3 + 0];
    av1.x = rowp[128 + kr * 3 + 1];  av1.y = rowp[128 + (kr + 1) * 3 + 1];
    av2.x = rowp[128 + kr * 3 + 2];  av2.y = rowp[128 + (kr + 1) * 3 + 2];
    // B fragments (weights L2/WGP$-resident)
    v2f bs, b0, b1;
    bs.x = Wsc[kr * 128 + col];  bs.y = Wsc[(kr + 1) * 128 + col];
    b0.x = Wu0[kr * 128 + col];  b0.y = Wu0[(kr + 1) * 128 + col];
    b1.x = Wu1[kr * 128 + col];  b1.y = Wu1[(kr + 1) * 128 + col];

    a_ns = wmma4(as,  bs, a_ns);
    a_su = wmma4(as,  b0, a_su);
    a_v0 = wmma4(av0, b1, a_v0);
    a_v1 = wmma4(av1, b1, a_v1);
    a_v2 = wmma4(av2, b1, a_v2);
  }

  const float inv = 0.08838834764831845f;  // 1/sqrt(128)
  #pragma unroll
  for (int v = 0; v < 8; ++v) {
    const int n = g * 16 + v + 8 * half;
    ns [n * 128 + col] = a_ns[v] * inv;
    sup[n * 128 + col] = a_su[v] * inv;
    vup[n * 384 + col * 3 + 0] = a_v0[v] * inv;
    vup[n * 384 + col * 3 + 1] = a_v1[v] * inv;
    vup[n * 384 + col * 3 + 2] = a_v2[v] * inv;
  }
}

// ---------------------------------------------------------------------------
// Kernel 2: fused edge pipeline.  grid = NE/16 blocks (16 edges each).
// Wave w computes columns [w*16, w*16+16) of every stage; for tpw it computes
// that column band of all four weight groups w0..w3 (shared A fragment).
// ---------------------------------------------------------------------------
__global__ __launch_bounds__(256)
void k_edge(const float* __restrict__ ea,   // edge_attrs  (NE,4)
            const float* __restrict__ ef,   // edge_feats  (NE,8)
            const float* __restrict__ len,  // lengths     (NE,1)
            const int*   __restrict__ ei,   // edge_index  (2,NE)
            const float* __restrict__ ns,   // node_scalars
            const float* __restrict__ sup,  // s_up
            const float* __restrict__ vup,  // v_up
            const float* __restrict__ W1, const float* __restrict__ b1,
            const float* __restrict__ W2, const float* __restrict__ b2,
            const float* __restrict__ W3,
            float* __restrict__ msg) {
  __shared__ float s_in[16 * MLPS];
  __shared__ float s_h1[16 * HSTR];
  __shared__ float s_h2[16 * HSTR];

  const int tid = threadIdx.x;
  const int e0  = blockIdx.x * 16;

  // ---- stage mlp_in: [ns[snd] | ns[rcv] | edge_feats | length | 0-pad] ----
  for (int i = tid; i < 16 * (MLPS / 4); i += 256) {
    const int row = i / (MLPS / 4), c = (i % (MLPS / 4)) * 4;
    const int e = e0 + row;
    float4 v = {0.f, 0.f, 0.f, 0.f};
    if (c < 128) {
      v = *(const float4*)(ns + (size_t)ei[e] * 128 + c);
    } else if (c < 256) {
      v = *(const float4*)(ns + (size_t)ei[NE + e] * 128 + (c - 128));
    } else if (c < 264) {
      v = *(const float4*)(ef + (size_t)e * 8 + (c - 256));
    } else if (c == 264) {
      v.x = len[e];
    }
    *(float4*)&s_in[row * MLPS + c] = v;
  }
  __syncthreads();

  const int wv = tid >> 5, lane = tid & 31;
  const int half = lane >> 4, lm = lane & 15;
  const int col = wv * 16 + lm;

  // ---- layer 1: (16 x 268) x (268 x 128), rows >= 265 of W1 are zero ----
  {
    v8f acc = {};
    for (int k = 0; k < DPAD; k += 4) {
      const int kr = k + 2 * half;
      v2f a = *(const v2f*)&s_in[lm * MLPS + kr];
      v2f b;
      b.x = (kr     < DIN) ? W1[(size_t)kr       * 128 + col] : 0.f;
      b.y = (kr + 1 < DIN) ? W1[(size_t)(kr + 1) * 128 + col] : 0.f;
      acc = wmma4(a, b, acc);
    }
    const float bias = b1[col];
    #pragma unroll
    for (int v = 0; v < 8; ++v)
      s_h1[(v + 8 * half) * HSTR + col] = siluf(acc[v] + bias);
  }
  __syncthreads();

  // ---- layer 2: (16 x 128) x (128 x 128) ----
  {
    v8f acc = {};
    for (int k = 0; k < 128; k += 4) {
      const int kr = k + 2 * half;
      v2f a = *(const v2f*)&s_h1[lm * HSTR + kr];
      v2f b;
      b.x = W2[kr * 128 + col];  b.y = W2[(kr + 1) * 128 + col];
      acc = wmma4(a, b, acc);
    }
    const float bias = b2[col];
    #pragma unroll
    for (int v = 0; v < 8; ++v)
      s_h2[(v + 8 * half) * HSTR + col] = siluf(acc[v] + bias);
  }
  __syncthreads();

  // ---- tpw: (16 x 128) x (128 x 512); wave owns col band of w0..w3 ----
  v8f t0 = {}, t1 = {}, t2 = {}, t3 = {};
  for (int k = 0; k < 128; k += 4) {
    const int kr = k + 2 * half;
    v2f a = *(const v2f*)&s_h2[lm * HSTR + kr];
    const float* r0 = W3 + (size_t)kr * 512 + col;
    const float* r1 = W3 + (size_t)(kr + 1) * 512 + col;
    v2f b0, b1_, b2_, b3_;
    b0.x = r0[0];    b0.y = r1[0];
    b1_.x = r0[128]; b1_.y = r1[128];
    b2_.x = r0[256]; b2_.y = r1[256];
    b3_.x = r0[384]; b3_.y = r1[384];
    t0 = wmma4(a, b0,  t0);   // shared A fragment across 4 WMMAs
    t1 = wmma4(a, b1_, t1);
    t2 = wmma4(a, b2_, t2);
    t3 = wmma4(a, b3_, t3);
  }

  // ---- tensor-product messages + segment-sum scatter (L2 atomics) ----
  #pragma unroll
  for (int v = 0; v < 8; ++v) {
    const int e = e0 + v + 8 * half;
    const int u = col;
    const int snd = ei[e], rcv = ei[NE + e];
    const float se = sup[(size_t)snd * 128 + u];
    const float* vp = vup + (size_t)snd * 384 + u * 3;
    const float ve0 = vp[0], ve1 = vp[1], ve2 = vp[2];
    const float* ap = ea + (size_t)e * 4;
    const float y0 = ap[0], y10 = ap[1], y11 = ap[2], y12 = ap[3];
    const float w0v = t0[v], w1v = t1[v], w2v = t2[v], w3v = t3[v];
    float* mr = msg + (size_t)rcv * 1024;

    atomicAdd(mr + u,       w0v * se * y0);
    atomicAdd(mr + 128 + u, w3v * (ve0 * y10 + ve1 * y11 + ve2 * y12) * INV_SQRT3);
    const float w1se = w1v * se;
    const float w2y0 = w2v * y0;
    atomicAdd(mr + 256 + u * 3 + 0, w1se * y10);
    atomicAdd(mr + 256 + u * 3 + 1, w1se * y11);
    atomicAdd(mr + 256 + u * 3 + 2, w1se * y12);
    atomicAdd(mr + 640 + u * 3 + 0, w2y0 * ve0);
    atomicAdd(mr + 640 + u * 3 + 1, w2y0 * ve1);
    atomicAdd(mr + 640 + u * 3 + 2, w2y0 * ve2);
  }
}

// ---------------------------------------------------------------------------
// Kernel 3: output GEMMs.  grid = NN/16.  msg rows read from L2/WGP$ directly
// (64 KB tile would exceed the static-LDS budget); 3 accumulators share the
// W_out1 B fragment.  Fused float4 store in the (n, 128, 4) interleave.
// ---------------------------------------------------------------------------
__global__ __launch_bounds__(256)
void k_out(const float* __restrict__ msg,
           const float* __restrict__ W0o,
           const float* __restrict__ W1o,
           float* __restrict__ out) {
  const int tid = threadIdx.x, g = blockIdx.x;
  const int wv = tid >> 5, lane = tid & 31;
  const int half = lane >> 4, lm = lane & 15;
  const int col = wv * 16 + lm;

  const float* arow = msg + (size_t)(g * 16 + lm) * 1024;

  v8f as_ = {}, av0 = {}, av1 = {}, av2 = {};
  for (int k = 0; k < 256; k += 4) {
    const int kr = k + 2 * half;
    v2f a_s = *(const v2f*)&arow[kr];
    v2f a0, a1, a2;
    a0.x = arow[256 + kr * 3 + 0];  a0.y = arow[256 + (kr + 1) * 3 + 0];
    a1.x = arow[256 + kr * 3 + 1];  a1.y = arow[256 + (kr + 1) * 3 + 1];
    a2.x = arow[256 + kr * 3 + 2];  a2.y = arow[256 + (kr + 1) * 3 + 2];
    v2f bs, bv;
    bs.x = W0o[kr * 128 + col];  bs.y = W0o[(kr + 1) * 128 + col];
    bv.x = W1o[kr * 128 + col];  bv.y = W1o[(kr + 1) * 128 + col];
    as_ = wmma4(a_s, bs, as_);
    av0 = wmma4(a0,  bv, av0);   // shared B fragment across 3 WMMAs
    av1 = wmma4(a1,  bv, av1);
    av2 = wmma4(a2,  bv, av2);
  }

  const float sc = 1.0f / 256.0f;  // (1/sqrt(256)) / AVG_NEIGH
  #pragma unroll
  for (int v = 0; v < 8; ++v) {
    const int n = g * 16 + v + 8 * half;
    float4 o = {as_[v] * sc, av0[v] * sc, av1[v] * sc, av2[v] * sc};
    *(float4*)(out + (size_t)n * 512 + col * 4) = o;
  }
}

// ---------------------------------------------------------------------------
extern "C" void kernel_launch(void* const* d_in, const int* in_sizes, int n_in,
                              void* d_out, int out_size, void* d_ws,
                              size_t ws_size, hipStream_t stream) {
  (void)in_sizes; (void)n_in; (void)out_size; (void)ws_size;
  const float* nf  = (const float*)d_in[0];
  const float* ea  = (const float*)d_in[1];
  const float* ef  = (const float*)d_in[2];
  const float* len = (const float*)d_in[3];
  const int*   ei  = (const int*)  d_in[4];
  const float* Wsc = (const float*)d_in[5];
  const float* Wu0 = (const float*)d_in[6];
  const float* Wu1 = (const float*)d_in[7];
  const float* W1  = (const float*)d_in[8];
  const float* b1  = (const float*)d_in[9];
  const float* W2  = (const float*)d_in[10];
  const float* b2  = (const float*)d_in[11];
  const float* W3  = (const float*)d_in[12];
  const float* W0o = (const float*)d_in[13];
  const float* W1o = (const float*)d_in[14];

  float* ws  = (float*)d_ws;
  float* ns  = ws;
  float* sup = ns  + (size_t)NN * 128;
  float* vup = sup + (size_t)NN * 128;
  float* msg = vup + (size_t)NN * 384;

  hipMemsetAsync(msg, 0, (size_t)NN * 1024 * sizeof(float), stream);
  k_node_up<<<NN / 16, 256, 0, stream>>>(nf, Wsc, Wu0, Wu1, ns, sup, vup);
  k_edge<<<NE / 16, 256, 0, stream>>>(ea, ef, len, ei, ns, sup, vup,
                                      W1, b1, W2, b2, W3, msg);
  k_out<<<NN / 16, 256, 0, stream>>>(msg, W0o, W1o, (float*)d_out);
}